// CausalMixer_21311627723306
// MI455X (gfx1250) — compile-verified
//
#include <hip/hip_runtime.h>
#include <math.h>

typedef __attribute__((ext_vector_type(2))) float v2f;
typedef __attribute__((ext_vector_type(8))) float v8f;
typedef __attribute__((ext_vector_type(4))) unsigned int v4u;
typedef __attribute__((ext_vector_type(8))) int v8i;
typedef __attribute__((ext_vector_type(4))) int v4i;

#define BB   128
#define TT   64
#define NA   10
#define NV   16
#define KK   4
#define SD   512
#define HH   256
#define EE   32
#define ROWS 32            // rows per block (2 x 16-row WMMA tiles)
#define NC1  848           // 844 used stage-1 columns, padded to x16
#define NT1  (NC1 / 16)    // 53 stage-1 column tiles
#define K8S1 (SD / 8)      // 64 packed k-groups, stage 1
#define NT2  34            // w1_l2 column tiles (544/16)
#define NT3  2             // w2_l2 column tiles (32/16)
#define K8S2 (HH / 8)      // 32 packed k-groups, stage 2

// column offsets inside the fused stage-1 output panel H1
#define C_HBASE 0     // 256: w00_l1 pre-activation
#define C_G     256   // 1:   b00 (g_base)
#define C_W01   257   // 10:  w01
#define C_B01   267   // 1:   b01
#define C_H1    268   // 256: w1_l1 pre-activation
#define C_B1    524   // 32:  b1
#define C_H2    556   // 256: w2_l1 pre-activation
#define C_HB2   812   // 32:  b2_l1 pre-activation

static __device__ __forceinline__ v8f wmma_f32_k4(v2f a, v2f b, v8f c) {
    // D = A(16x4 f32) * B(4x16 f32) + C(16x16 f32)
    return __builtin_amdgcn_wmma_f32_16x16x4_f32(
        /*neg_a=*/false, a, /*neg_b=*/false, b,
        /*c_mod=*/(short)0, c, /*reuse_a=*/false, /*reuse_b=*/false);
}

// logical concatenated stage-1 weight matrix element W[k][n]
static __device__ __forceinline__ float colval(
    int k, int n,
    const float* __restrict__ w00_l1_W, const float* __restrict__ b00_W,
    const float* __restrict__ w01_W,    const float* __restrict__ b01_W,
    const float* __restrict__ w1_l1_W,  const float* __restrict__ b1_W,
    const float* __restrict__ w2_l1_W,  const float* __restrict__ b2_l1_W)
{
    if      (n < 256)  return w00_l1_W[k * 256 + n];
    else if (n == 256) return b00_W[k];
    else if (n < 267)  return w01_W[k * 10 + (n - 257)];
    else if (n == 267) return b01_W[k];
    else if (n < 524)  return w1_l1_W[k * 256 + (n - 268)];
    else if (n < 556)  return b1_W[k * 32 + (n - 524)];
    else if (n < 812)  return w2_l1_W[k * 256 + (n - 556)];
    else if (n < 844)  return b2_l1_W[k * 32 + (n - 812)];
    return 0.0f;
}

// decode fragment-linear index -> (k, n) of the B element a lane owns
// layout: Wp[((n0*K8 + k8)*32 + lane)*4 + j]
static __device__ __forceinline__ void frag_decode(long long idx, int K8,
                                                   int& k, int& n) {
    const int j    = (int)(idx & 3);
    const int lane = (int)((idx >> 2) & 31);
    const long long rest = idx >> 7;
    const int k8 = (int)(rest % K8);
    const int n0 = (int)(rest / K8);
    k = k8 * 8 + ((j >> 1) << 2) + ((lane >> 4) << 1) + (j & 1);
    n = n0 * 16 + (lane & 15);
}

// ---------------- prep: pack stage-1 B into fragment-linear layout ----------
__global__ void pack_stage1(
    const float* __restrict__ w00_l1_W, const float* __restrict__ b00_W,
    const float* __restrict__ w01_W,    const float* __restrict__ b01_W,
    const float* __restrict__ w1_l1_W,  const float* __restrict__ b1_W,
    const float* __restrict__ w2_l1_W,  const float* __restrict__ b2_l1_W,
    float* __restrict__ Wp1)
{
    const long long total = (long long)NT1 * K8S1 * 128;
    for (long long idx = blockIdx.x * (long long)blockDim.x + threadIdx.x;
         idx < total; idx += (long long)gridDim.x * blockDim.x) {
        int k, n;
        frag_decode(idx, K8S1, k, n);
        Wp1[idx] = colval(k, n, w00_l1_W, b00_W, w01_W, b01_W,
                          w1_l1_W, b1_W, w2_l1_W, b2_l1_W);
    }
}

// pack a plain row-major (K x N) matrix into fragment-linear layout
__global__ void pack_rowmajor(const float* __restrict__ W, int N, int K8,
                              float* __restrict__ Wp)
{
    const long long total = (long long)(N / 16) * K8 * 128;
    for (long long idx = blockIdx.x * (long long)blockDim.x + threadIdx.x;
         idx < total; idx += (long long)gridDim.x * blockDim.x) {
        int k, n;
        frag_decode(idx, K8, k, n);
        Wp[idx] = W[k * N + n];
    }
}

__global__ void prep_misc(
    const float* __restrict__ w00_l1_W, const float* __restrict__ w00_l1_b,
    const float* __restrict__ b00_W,    const float* __restrict__ b00_b,
    const float* __restrict__ w01_b,    const float* __restrict__ b01_b,
    const float* __restrict__ w1_l1_b,  const float* __restrict__ b1_b,
    const float* __restrict__ w2_l1_b,  const float* __restrict__ b2_l1_b,
    float* __restrict__ bcat, float* __restrict__ h_delta, float* __restrict__ g_delta)
{
    const int tid = threadIdx.x;
    for (int n = tid; n < NC1; n += blockDim.x) {
        float v;
        if      (n < 256)  v = w00_l1_b[n];
        else if (n == 256) v = b00_b[0];
        else if (n < 267)  v = w01_b[n - 257];
        else if (n == 267) v = b01_b[0];
        else if (n < 524)  v = w1_l1_b[n - 268];
        else if (n < 556)  v = b1_b[n - 524];
        else if (n < 812)  v = w2_l1_b[n - 556];
        else if (n < 844)  v = b2_l1_b[n - 812];
        else               v = 0.0f;
        bcat[n] = v;
    }
    if (tid < 256) {                          // all-ones one-hot row contribution
        float s = 0.0f;
        for (int r = 0; r < NV; ++r) s += w00_l1_W[(SD + r) * 256 + tid];
        h_delta[tid] = s;
    }
    if (tid == 0) {
        float s = 0.0f;
        for (int r = 0; r < NV; ++r) s += b00_W[SD + r];
        g_delta[0] = s;
    }
}

// ---------------- fused main kernel -----------------------------------------
__global__ __launch_bounds__(256)
void causal_mixer_main(
    const float* __restrict__ qvals, const int* __restrict__ crel,
    const float* __restrict__ states,
    const float* __restrict__ Wp1, const float* __restrict__ Wp2,
    const float* __restrict__ Wp3, const float* __restrict__ bcat,
    const float* __restrict__ h_delta, const float* __restrict__ g_delta,
    const float* __restrict__ w00_l2_W, const float* __restrict__ w00_l2_b,
    const float* __restrict__ w1_l2_b,  const float* __restrict__ w2_l2_b,
    const float* __restrict__ b2_l2_W,  const float* __restrict__ b2_l2_b,
    float* __restrict__ out)
{
    extern __shared__ float smem[];
    float* sS   = smem;                 // 32 x 512  states tile
    float* sH1  = sS  + ROWS * SD;      // 32 x 848  stage-1 panel
    float* sW1  = sH1 + ROWS * NC1;     // 32 x 544  |w1|
    float* sW2  = sW1 + ROWS * 544;     // 32 x 32   |w2|
    float* sW0s = sW2 + ROWS * 32;      // 32 x 4    w00 (shared variant)
    float* sW0d = sW0s + ROWS * 4;      // 32 x 4    w00 (v==b variant)
    float* sB2  = sW0d + ROWS * 4;      // 32        b2

    const int tid  = threadIdx.x;
    const int lane = tid & 31;
    const int w    = tid >> 5;                 // wave id 0..7 (uniform per wave)
    const int g0   = blockIdx.x * ROWS;        // first global (b,t) row
    const int bb   = g0 / TT;                  // batch index, constant per block

    // ---- stage states tile into LDS via Tensor Data Mover (wave 0 issues) ----
    if (tid < 32) {
        const unsigned long long ga =
            (unsigned long long)(const void*)(states + (size_t)g0 * SD);
        const unsigned int lds = (unsigned int)(unsigned long long)(void*)sS;
        const unsigned int td0 = ROWS * SD;    // 16384 f32 elements, contiguous

        // D# group 0: count=1 | lds_addr | global_addr[56:0] | type=2
        v4u d0;
        d0.x = 1u;                                        // count=1, user mode
        d0.y = lds;                                       // lds_addr (bytes)
        d0.z = (unsigned int)ga;                          // global_addr[31:0]
        d0.w = ((unsigned int)(ga >> 32) & 0x01FFFFFFu)   // global_addr[56:32]
               | 0x80000000u;                             // type=2 (image)

        // D# group 1: mask=0, data_size=4B, tensor_dim0=tile_dim0=16384,
        //             tensor_dim1=tile_dim1=1, dim0_stride=16384
        v8i d1;
        d1[0] = (int)(2u << 16);                          // data_size = 4B
        d1[1] = (int)((td0 & 0xFFFFu) << 16);             // tensor_dim0[15:0]
        d1[2] = (int)((td0 >> 16) | (1u << 16));          // t_dim0[31:16] | t_dim1[15:0]=1
        d1[3] = (int)((td0 & 0xFFFFu) << 16);             // tile_dim0 = 16384
        d1[4] = 1;                                        // tile_dim1 = 1
        d1[5] = (int)td0;                                 // dim0_stride[31:0]
        d1[6] = 0;                                        // dim0_stride[47:32], dim1_stride lo
        d1[7] = 0;

        const v4i dz4 = {0, 0, 0, 0};                     // groups 2/3 unused (<=2D)
        const v8i dz8 = {0, 0, 0, 0, 0, 0, 0, 0};
        __builtin_amdgcn_tensor_load_to_lds(d0, d1, dz4, dz4, dz8, 0);
        __builtin_amdgcn_s_wait_tensorcnt(0);
    }
    __syncthreads();

    const int mr = lane & 15;                  // row (A) / col (B,C) within tile
    const int kh = (lane >> 4) << 1;           // 0 or 2: K-subslot for half-wave
    const int rb = (lane >> 4) << 3;           // 0 or 8: C row base

    // ---- stage 1: sH1 = sS @ Wcat + bcat; B fragment reused across 2 A tiles
    for (int n0 = w; n0 < NT1; n0 += 8) {
        const int nc = n0 * 16 + mr;
        const float4* bp = (const float4*)Wp1 + (size_t)n0 * K8S1 * 32 + lane;
        const float* a0p = sS + mr * SD;             // row tile 0
        const float* a1p = sS + (16 + mr) * SD;      // row tile 1
        v8f acc0 = {}, acc1 = {};
#pragma unroll 4
        for (int k8 = 0; k8 < K8S1; ++k8) {
            const float4 bq = bp[(size_t)k8 * 32];   // coalesced b128
            const v2f b0 = {bq.x, bq.y};
            const v2f b1 = {bq.z, bq.w};
            const int k0 = k8 * 8 + kh;
            v2f a;
            a.x = a0p[k0];     a.y = a0p[k0 + 1];
            acc0 = wmma_f32_k4(a, b0, acc0);
            a.x = a1p[k0];     a.y = a1p[k0 + 1];
            acc1 = wmma_f32_k4(a, b0, acc1);
            a.x = a0p[k0 + 4]; a.y = a0p[k0 + 5];
            acc0 = wmma_f32_k4(a, b1, acc0);
            a.x = a1p[k0 + 4]; a.y = a1p[k0 + 5];
            acc1 = wmma_f32_k4(a, b1, acc1);
        }
        const float bias = bcat[nc];
#pragma unroll
        for (int r = 0; r < 8; ++r) {
            sH1[(rb + r) * NC1 + nc]        = acc0[r] + bias;
            sH1[(16 + rb + r) * NC1 + nc]   = acc1[r] + bias;
        }
    }
    __syncthreads();

    // ---- stage 2: |relu(h1)@w1_l2 + b| (34 tiles) and |relu(h2)@w2_l2 + b| (2)
    for (int t = w; t < NT2 + NT3; t += 8) {   // t uniform per wave
        int n0, aoff, ldD;
        const float *Pp, *bp; float* Dp;
        if (t < NT2) { n0 = t;       aoff = C_H1; Pp = Wp2; bp = w1_l2_b; Dp = sW1; ldD = 544; }
        else         { n0 = t - NT2; aoff = C_H2; Pp = Wp3; bp = w2_l2_b; Dp = sW2; ldD = 32;  }
        const int nc = n0 * 16 + mr;
        const float4* bpq = (const float4*)Pp + (size_t)n0 * K8S2 * 32 + lane;
        const float* a0p = sH1 + mr * NC1 + aoff;
        const float* a1p = sH1 + (16 + mr) * NC1 + aoff;
        v8f acc0 = {}, acc1 = {};
#pragma unroll 4
        for (int k8 = 0; k8 < K8S2; ++k8) {
            const float4 bq = bpq[(size_t)k8 * 32];
            const v2f b0 = {bq.x, bq.y};
            const v2f b1 = {bq.z, bq.w};
            const int k0 = k8 * 8 + kh;
            v2f a;
            a.x = fmaxf(a0p[k0], 0.f);     a.y = fmaxf(a0p[k0 + 1], 0.f);
            acc0 = wmma_f32_k4(a, b0, acc0);
            a.x = fmaxf(a1p[k0], 0.f);     a.y = fmaxf(a1p[k0 + 1], 0.f);
            acc1 = wmma_f32_k4(a, b0, acc1);
            a.x = fmaxf(a0p[k0 + 4], 0.f); a.y = fmaxf(a0p[k0 + 5], 0.f);
            acc0 = wmma_f32_k4(a, b1, acc0);
            a.x = fmaxf(a1p[k0 + 4], 0.f); a.y = fmaxf(a1p[k0 + 5], 0.f);
            acc1 = wmma_f32_k4(a, b1, acc1);
        }
        const float bias = bp[nc];
#pragma unroll
        for (int r = 0; r < 8; ++r) {
            Dp[(rb + r) * ldD + nc]      = fabsf(acc0[r] + bias);
            Dp[(16 + rb + r) * ldD + nc] = fabsf(acc1[r] + bias);
        }
    }

    // ---- tiny dots on VALU (read-only on sH1, disjoint writes) ----
    if (tid < ROWS * 4) {                      // w00_l2: 32 rows x 4 cols, K=256
        const int m = tid >> 2, j = tid & 3;
        float s  = w00_l2_b[j];
        float sd = s;
        for (int i = 0; i < HH; ++i) {
            const float h  = sH1[m * NC1 + C_HBASE + i];
            const float wv = w00_l2_W[i * 4 + j];
            s  += fmaxf(h, 0.0f) * wv;
            sd += fmaxf(h + h_delta[i], 0.0f) * wv;
        }
        sW0s[tid] = s;
        sW0d[tid] = sd;
    }
    if (tid >= 128 && tid < 128 + ROWS) {      // b2 head: K=32
        const int m = tid - 128;
        float s = b2_l2_b[0];
        for (int e = 0; e < EE; ++e)
            s += fmaxf(sH1[m * NC1 + C_HB2 + e], 0.0f) * b2_l2_W[e];
        sB2[m] = s;
    }
    __syncthreads();

    // ---- per-row epilogue: gather, group mix, elu mix, output ----
    if (tid < ROWS) {
        const int m = tid;
        const int g = g0 + m;

        float qv[NA];
#pragma unroll
        for (int a = 0; a < NA; ++a) qv[a] = qvals[g * NA + a];
        const int* crp = crel + (size_t)g * KK * NV;

        float gq[NV + 1];
        const float gbase = sH1[m * NC1 + C_G];
        const float gdel  = g_delta[0];
        for (int v = 0; v < NV; ++v) {
            const float* w0row = (bb == v) ? (sW0d + m * 4) : (sW0s + m * 4);
            float s = gbase + ((bb == v) ? gdel : 0.0f);
#pragma unroll
            for (int k = 0; k < KK; ++k) {
                const int a = crp[k * NV + v];
                s += qv[a] * fabsf(w0row[k]);
            }
            gq[v] = s;
        }
        float other = sH1[m * NC1 + C_B01];
#pragma unroll
        for (int a = 0; a < NA; ++a) other += qv[a] * sH1[m * NC1 + C_W01 + a];
        gq[NV] = other;

        float y = sB2[m];
        for (int e = 0; e < EE; ++e) {
            float h = sH1[m * NC1 + C_B1 + e];
#pragma unroll
            for (int v = 0; v <= NV; ++v)
                h += gq[v] * sW1[m * 544 + v * EE + e];
            const float hid = (h > 0.0f) ? h : (expf(h) - 1.0f);
            y += hid * sW2[m * EE + e];
        }
        out[g] = y;
    }
}

// ---------------- host launcher ---------------------------------------------
extern "C" void kernel_launch(void* const* d_in, const int* in_sizes, int n_in,
                              void* d_out, int out_size, void* d_ws, size_t ws_size,
                              hipStream_t stream)
{
    (void)in_sizes; (void)n_in; (void)out_size; (void)ws_size;

    const float* qvals    = (const float*)d_in[0];
    const int*   crel     = (const int*)  d_in[1];
    const float* states   = (const float*)d_in[2];
    const float* w00_l1_W = (const float*)d_in[3];
    const float* w00_l1_b = (const float*)d_in[4];
    const float* w00_l2_W = (const float*)d_in[5];
    const float* w00_l2_b = (const float*)d_in[6];
    const float* b00_W    = (const float*)d_in[7];
    const float* b00_b    = (const float*)d_in[8];
    const float* w01_W    = (const float*)d_in[9];
    const float* w01_b    = (const float*)d_in[10];
    const float* b01_W    = (const float*)d_in[11];
    const float* b01_b    = (const float*)d_in[12];
    const float* w1_l1_W  = (const float*)d_in[13];
    const float* w1_l1_b  = (const float*)d_in[14];
    const float* w1_l2_W  = (const float*)d_in[15];
    const float* w1_l2_b  = (const float*)d_in[16];
    const float* b1_W     = (const float*)d_in[17];
    const float* b1_b     = (const float*)d_in[18];
    const float* w2_l1_W  = (const float*)d_in[19];
    const float* w2_l1_b  = (const float*)d_in[20];
    const float* w2_l2_W  = (const float*)d_in[21];
    const float* w2_l2_b  = (const float*)d_in[22];
    const float* b2_l1_W  = (const float*)d_in[23];
    const float* b2_l1_b  = (const float*)d_in[24];
    const float* b2_l2_W  = (const float*)d_in[25];
    const float* b2_l2_b  = (const float*)d_in[26];

    // workspace layout (floats)
    float* Wp1     = (float*)d_ws;                         // 53*64*128 = 434176
    float* Wp2     = Wp1 + (size_t)NT1 * K8S1 * 128;       // 34*32*128 = 139264
    float* Wp3     = Wp2 + (size_t)NT2 * K8S2 * 128;       // 2*32*128  = 8192
    float* bcat    = Wp3 + (size_t)NT3 * K8S2 * 128;       // 848
    float* h_delta = bcat + NC1;                           // 256
    float* g_delta = h_delta + 256;                        // 1

    pack_stage1<<<1696, 256, 0, stream>>>(w00_l1_W, b00_W, w01_W, b01_W,
                                          w1_l1_W, b1_W, w2_l1_W, b2_l1_W, Wp1);
    pack_rowmajor<<<544, 256, 0, stream>>>(w1_l2_W, 544, K8S2, Wp2);
    pack_rowmajor<<<32, 256, 0, stream>>>(w2_l2_W, 32, K8S2, Wp3);
    prep_misc<<<1, 256, 0, stream>>>(w00_l1_W, w00_l1_b, b00_W, b00_b,
                                     w01_b, b01_b, w1_l1_b, b1_b,
                                     w2_l1_b, b2_l1_b, bcat, h_delta, g_delta);

    const int nBlocks = (BB * TT) / ROWS;                  // 256
    const size_t shmem =
        (size_t)(ROWS * SD + ROWS * NC1 + ROWS * 544 + ROWS * 32 +
                 ROWS * 4 + ROWS * 4 + ROWS) * sizeof(float);  // ~243 KB
    causal_mixer_main<<<nBlocks, 256, shmem, stream>>>(
        qvals, crel, states, Wp1, Wp2, Wp3, bcat, h_delta, g_delta,
        w00_l2_W, w00_l2_b, w1_l2_b, w2_l2_b,
        b2_l2_W, b2_l2_b, (float*)d_out);
}